// RGCNLayer_7138235646652
// MI455X (gfx1250) — compile-verified
//
#include <hip/hip_runtime.h>

typedef __attribute__((ext_vector_type(2))) float v2f;
typedef __attribute__((ext_vector_type(4))) float v4f;
typedef __attribute__((ext_vector_type(8))) float v8f;

#define IN_FEAT   128
#define OUT_FEAT  128
#define NUM_BASES 4
#define YCOLS     (NUM_BASES * OUT_FEAT)   // 512 floats per node row of y

// ---------------------------------------------------------------------------
// Phase 1: y[n][c*16 + j] = sum_i x[n][i] * W_bases[b][i][o],  b=c/8, o=(c%8)*16+j
//   Dense f32 GEMM [N,128] @ [128, 4*128] using V_WMMA_F32_16X16X4_F32.
//   One wave computes one 16(node) x 16(col) tile; 8 waves/block cover 8 column
//   tiles; blockIdx.y covers the 32 column tiles, blockIdx.x the node tiles.
//   Fragment layouts follow CDNA5 ISA §7.12.2:
//     A 16x4 f32 : lanes 0-15 -> M, VGPR0 = K {0|2}, VGPR1 = K {1|3}
//     B 4x16 f32 : lanes 0-15 -> N, VGPR0 = K {0|2}, VGPR1 = K {1|3}
//     C 16x16 f32: VGPR r, lanes 0-15 -> (M=r, N=lane); lanes 16-31 -> (M=r+8)
// ---------------------------------------------------------------------------
__global__ __launch_bounds__(256) void rgcn_xw_wmma(
    const float* __restrict__ x,      // [N, 128]
    const float* __restrict__ Wb,     // [4, 128, 128]
    float* __restrict__ y,            // [N, 512]
    int n_nodes) {
  const int lane = threadIdx.x & 31;
  const int wave = threadIdx.x >> 5;
  const int half = lane >> 4;          // 0: lanes 0-15, 1: lanes 16-31
  const int m    = lane & 15;          // M index (A), N index (B/C)
  const int c    = (blockIdx.y << 3) + wave;   // column tile 0..31
  const int n0   = blockIdx.x << 4;            // node tile base
  if (n0 >= n_nodes) return;

  const int b  = c >> 3;               // basis index 0..3
  const int o0 = (c & 7) << 4;         // output-feature base within basis

  // A: lane holds row (n0+m); pairs of K values -> contiguous float2 loads
  const float* xrow  = x + (size_t)(n0 + m) * IN_FEAT + 2 * half;
  // B: lane holds column (o0+m); K advances down rows of W (stride 128)
  const float* wbase = Wb + (size_t)b * IN_FEAT * OUT_FEAT + o0 + m;

  v8f acc = {};
#pragma unroll
  for (int k0 = 0; k0 < IN_FEAT; k0 += 4) {
    v2f a = *(const v2f*)(xrow + k0);                       // K = k0+2h, k0+2h+1
    v2f bb;
    bb.x = wbase[(size_t)(k0 + 2 * half) * OUT_FEAT];       // K = k0+2h
    bb.y = wbase[(size_t)(k0 + 2 * half + 1) * OUT_FEAT];   // K = k0+2h+1
    // (neg_a, A, neg_b, B, c_mod, C, reuse_a, reuse_b)
    acc = __builtin_amdgcn_wmma_f32_16x16x4_f32(
        false, a, false, bb, (short)0, acc, false, false);
  }

  // Store C tile: row (n0 + r + 8*half), column c*16 + m of y[N,512]
  float* yout = y + (size_t)n0 * YCOLS + c * 16 + m;
#pragma unroll
  for (int r = 0; r < 8; ++r) {
    yout[(size_t)(r + 8 * half) * YCOLS] = acc[r];
  }
}

// ---------------------------------------------------------------------------
// Phase 2: out[dst_e] += norm_e * sum_b w_comp[etype_e][b] * y[src_e][b,:]
//   One wave per edge. Edge scalars are wave-uniform -> SGPRs via
//   readfirstlane (saddr-form loads/atomics). Lane handles 4 consecutive
//   output features: 4x float4 gathers from the L2-resident y table, FMA
//   combine, then 4 hardware f32 atomic adds into the L2-resident output.
// ---------------------------------------------------------------------------
__global__ __launch_bounds__(256) void rgcn_edge_scatter(
    const float* __restrict__ y,      // [N, 512]
    const float* __restrict__ wcomp,  // [16, 4]
    const float* __restrict__ norm,   // [E]
    const int*   __restrict__ src,    // [E]
    const int*   __restrict__ dst,    // [E]
    const int*   __restrict__ etype,  // [E]
    float* __restrict__ out,          // [N, 128]
    int n_edges) {
  const int lane = threadIdx.x & 31;
  const int wave = threadIdx.x >> 5;
  const int e = blockIdx.x * 8 + wave;
  if (e >= n_edges) return;

  const int s = __builtin_amdgcn_readfirstlane(src[e]);
  const int d = __builtin_amdgcn_readfirstlane(dst[e]);
  const int t = __builtin_amdgcn_readfirstlane(etype[e]);
  const float nv = norm[e];

  const float c0 = nv * wcomp[t * 4 + 0];
  const float c1 = nv * wcomp[t * 4 + 1];
  const float c2 = nv * wcomp[t * 4 + 2];
  const float c3 = nv * wcomp[t * 4 + 3];

  const v4f* yv = (const v4f*)(y + (size_t)s * YCOLS);  // 128 x float4 per node
  v4f acc = yv[lane] * c0;            // basis 0, features lane*4..lane*4+3
  acc += yv[32 + lane] * c1;          // basis 1
  acc += yv[64 + lane] * c2;          // basis 2
  acc += yv[96 + lane] * c3;          // basis 3

  float* op = out + (size_t)d * OUT_FEAT + lane * 4;
  unsafeAtomicAdd(op + 0, acc.x);
  unsafeAtomicAdd(op + 1, acc.y);
  unsafeAtomicAdd(op + 2, acc.z);
  unsafeAtomicAdd(op + 3, acc.w);
}

// ---------------------------------------------------------------------------
// Launch: zero out, dense WMMA transform into workspace, edge scatter.
// Workspace use: y = [n_nodes, 512] f32 (102.4 MB for N=50000; L2-resident).
// ---------------------------------------------------------------------------
extern "C" void kernel_launch(void* const* d_in, const int* in_sizes, int n_in,
                              void* d_out, int out_size, void* d_ws, size_t ws_size,
                              hipStream_t stream) {
  const float* x   = (const float*)d_in[0];  // [N,128]
  const float* Wb  = (const float*)d_in[1];  // [4,128,128]
  const float* wc  = (const float*)d_in[2];  // [16,4]
  const float* nrm = (const float*)d_in[3];  // [E,1]
  const int*   src = (const int*)d_in[4];    // [E]
  const int*   dst = (const int*)d_in[5];    // [E]
  const int*   ety = (const int*)d_in[6];    // [E]
  float* out = (float*)d_out;                // [N,128]
  float* y   = (float*)d_ws;                 // [N,512]

  const int n_nodes = in_sizes[0] / IN_FEAT;
  const int n_edges = in_sizes[4];

  hipMemsetAsync(out, 0, (size_t)out_size * sizeof(float), stream);

  dim3 g1((n_nodes + 15) / 16, 4);           // node tiles x (32 col tiles / 8 waves)
  rgcn_xw_wmma<<<g1, 256, 0, stream>>>(x, Wb, y, n_nodes);

  const int edges_per_block = 8;             // 8 waves/block, 1 edge/wave
  dim3 g2((n_edges + edges_per_block - 1) / edges_per_block);
  rgcn_edge_scatter<<<g2, 256, 0, stream>>>(y, wc, nrm, src, dst, ety, out, n_edges);
}